// RGCNLayer_70617852281332
// MI455X (gfx1250) — compile-verified
//
#include <hip/hip_runtime.h>

typedef __attribute__((ext_vector_type(2))) float v2f;
typedef __attribute__((ext_vector_type(4))) float v4f;
typedef __attribute__((ext_vector_type(8))) float v8f;

#define DIN   64
#define DOUT  64
#define NREL  8
#define LDSTR 68  /* col-major LDS stride: banks (4*col + k) % 64 conflict-free */

// ---------------------------------------------------------------- utilities
__global__ void zero_f32(float* __restrict__ p, size_t n) {
  size_t i = (size_t)blockIdx.x * blockDim.x + threadIdx.x;
  size_t stride = (size_t)gridDim.x * blockDim.x;
  for (; i < n; i += stride) p[i] = 0.0f;
}

__global__ void relu_inplace(float* __restrict__ p, size_t n) {
  size_t i = (size_t)blockIdx.x * blockDim.x + threadIdx.x;
  size_t stride = (size_t)gridDim.x * blockDim.x;
  for (; i < n; i += stride) p[i] = fmaxf(p[i], 0.0f);
}

// ------------------------------------------------- pass 1: scatter features
__global__ void scatter_feat(const float* __restrict__ x,
                             const int* __restrict__ edge_index,
                             const int* __restrict__ edge_type,
                             float* __restrict__ agg, float* __restrict__ cnt,
                             int n_edges) {
  size_t gid = (size_t)blockIdx.x * blockDim.x + threadIdx.x;
  int e = (int)(gid >> 6);
  int i = (int)(gid & 63);
  if (e >= n_edges) return;
  int src = edge_index[e];
  int dst = edge_index[n_edges + e];
  int r   = edge_type[e];
  atomicAdd(&agg[(size_t)dst * (NREL * DIN) + r * DIN + i],
            x[(size_t)src * DIN + i]);
  if (i == 0) atomicAdd(&cnt[(size_t)dst * NREL + r], 1.0f);
}

__global__ void count_edges(const int* __restrict__ edge_index,
                            const int* __restrict__ edge_type,
                            float* __restrict__ cnt, int n_edges) {
  int e = blockIdx.x * blockDim.x + threadIdx.x;
  if (e >= n_edges) return;
  int dst = edge_index[n_edges + e];
  atomicAdd(&cnt[(size_t)dst * NREL + edge_type[e]], 1.0f);
}

// ---------------------------------------- pass 2: fused WMMA GEMM + epilogue
// out[16 x 64 tile] = relu( [x | mean_0..mean_7] @ [Wroot; W_0..W_7] + bias )
// 9 compile-time K-segments of 64; each segment's 64x64 weight block is
// staged transposed into LDS and shared by the 8 waves of the block.
template <bool FULL>
__global__ __launch_bounds__(256) void rgcn_gemm(
    const float* __restrict__ x,      // [N][64]
    const float* __restrict__ W,      // [8][64][64]
    const float* __restrict__ Wroot,  // [64][64]
    const float* __restrict__ bias,   // [64]
    const float* __restrict__ agg,    // [N][8][64]  (FULL only)
    const float* __restrict__ cnt,    // [N][8]      (FULL only)
    float* __restrict__ out, int n_nodes, int n_tiles) {
  __shared__ float Bs[DOUT * LDSTR];  // 17 KB

  const int tid  = threadIdx.x;
  const int lane = tid & 31;
  const int wave = tid >> 5;
  int tile = blockIdx.x * 8 + wave;
  if (tile >= n_tiles) tile = n_tiles - 1;  // keep wave alive for barriers
  const int rowBase = tile * 16;

  const int m  = lane & 15;  // A row / B,C,D column within tile
  const int hi = lane >> 4;  // K pair {0,1} vs {2,3}; C/D row +8
  int row = rowBase + m;
  if (row >= n_nodes) row = n_nodes - 1;  // clamp loads; stores guarded

  float invc[NREL];
  if (FULL) {
#pragma unroll
    for (int r = 0; r < NREL; ++r)
      invc[r] = 1.0f / fmaxf(cnt[(size_t)row * NREL + r], 1.0f);
  }

  v8f acc[4] = {};

  const int nseg = FULL ? (1 + NREL) : 1;
#pragma unroll
  for (int s = 0; s < nseg; ++s) {
    const float* Bsrc = (s == 0) ? Wroot : (W + (size_t)(s - 1) * DIN * DOUT);

    __syncthreads();  // previous segment fully consumed before restage
    // stage 64x64 weight block, transposed to col-major (b128 global reads)
#pragma unroll
    for (int it = 0; it < 4; ++it) {
      const int idx = (it * 256 + tid) * 4;  // flat = kin*64 + col
      const int kin = idx >> 6;
      const int col = idx & 63;
      v4f v = *(const v4f*)(Bsrc + idx);
      Bs[(col + 0) * LDSTR + kin] = v.x;
      Bs[(col + 1) * LDSTR + kin] = v.y;
      Bs[(col + 2) * LDSTR + kin] = v.z;
      Bs[(col + 3) * LDSTR + kin] = v.w;
    }
    __syncthreads();

    const float* Arow =
        (s == 0) ? (x + (size_t)row * DIN)
                 : (agg + (size_t)row * (NREL * DIN) + (s - 1) * DIN);
    const float scale = (s == 0) ? 1.0f : invc[s - 1];  // static index (unrolled)

#pragma unroll
    for (int k0 = 0; k0 < DIN; k0 += 4) {
      const int ka = k0 + hi * 2;  // even -> 8B-aligned b64 loads
      v2f a = *(const v2f*)(Arow + ka);
      if (s != 0) { a.x *= scale; a.y *= scale; }
#pragma unroll
      for (int t = 0; t < 4; ++t) {
        const int col = t * 16 + m;
        v2f b = *(const v2f*)(&Bs[col * LDSTR + ka]);  // ds_load_b64, no conflicts
        acc[t] = __builtin_amdgcn_wmma_f32_16x16x4_f32(
            false, a, false, b, (short)0, acc[t], false, false);
      }
    }
  }

  // epilogue: +bias, relu (FULL only), guarded stores
#pragma unroll
  for (int t = 0; t < 4; ++t) {
    const int col = t * 16 + m;
    const float bv = bias[col];
#pragma unroll
    for (int j = 0; j < 8; ++j) {
      const int grow = rowBase + hi * 8 + j;  // C/D: VGPR j -> row j (+8 hi)
      if (grow < n_nodes) {
        float v = acc[t][j] + bv;
        out[(size_t)grow * DOUT + col] = FULL ? fmaxf(v, 0.0f) : v;
      }
    }
  }
}

// -------------------- fallback: per-edge transform with atomics into d_out
__global__ void edge_mm(const float* __restrict__ x,
                        const float* __restrict__ W,
                        const int* __restrict__ edge_index,
                        const int* __restrict__ edge_type,
                        const float* __restrict__ cnt,
                        float* __restrict__ out, int n_edges) {
  size_t gid = (size_t)blockIdx.x * blockDim.x + threadIdx.x;
  int e = (int)(gid >> 6);
  int i = (int)(gid & 63);
  if (e >= n_edges) return;
  int src = edge_index[e];
  int dst = edge_index[n_edges + e];
  int r   = edge_type[e];
  float s = 1.0f / fmaxf(cnt[(size_t)dst * NREL + r], 1.0f);
  const float* xs = x + (size_t)src * DIN;
  const float* wr = W + (size_t)r * DIN * DOUT;
  float sum = 0.0f;
#pragma unroll 8
  for (int k = 0; k < DIN; ++k) sum += xs[k] * wr[k * DOUT + i];
  atomicAdd(&out[(size_t)dst * DOUT + i], sum * s);
}

// ---------------------------------------------------------------- launcher
extern "C" void kernel_launch(void* const* d_in, const int* in_sizes, int n_in,
                              void* d_out, int out_size, void* d_ws,
                              size_t ws_size, hipStream_t stream) {
  const float* x    = (const float*)d_in[0];
  const int*   ei   = (const int*)d_in[1];
  const int*   et   = (const int*)d_in[2];
  const float* W    = (const float*)d_in[3];
  const float* Wr   = (const float*)d_in[4];
  const float* bias = (const float*)d_in[5];
  float* out = (float*)d_out;

  const int n_nodes = in_sizes[0] / DIN;
  const int n_edges = in_sizes[2];

  const size_t aggElems = (size_t)n_nodes * NREL * DIN;
  const size_t cntElems = (size_t)n_nodes * NREL;

  const int tiles = (n_nodes + 15) / 16;
  const int gemmBlocks = (tiles + 7) / 8;  // 8 waves / block (wave32)

  const size_t edgeThreads = (size_t)n_edges * 64;
  const int edgeBlocks = (int)((edgeThreads + 255) / 256);

  if (ws_size >= (aggElems + cntElems) * sizeof(float)) {
    // Primary path: scatter-aggregate, then one fused WMMA GEMM (K = 576).
    float* agg = (float*)d_ws;
    float* cnt = agg + aggElems;
    zero_f32<<<4096, 256, 0, stream>>>(agg, aggElems + cntElems);
    scatter_feat<<<edgeBlocks, 256, 0, stream>>>(x, ei, et, agg, cnt, n_edges);
    rgcn_gemm<true><<<gemmBlocks, 256, 0, stream>>>(x, W, Wr, bias, agg, cnt,
                                                    out, n_nodes, tiles);
  } else {
    // Small-scratch fallback: counts only; root GEMM on WMMA, edges via VALU.
    float* cnt = (float*)d_ws;
    zero_f32<<<1024, 256, 0, stream>>>(cnt, cntElems);
    count_edges<<<(n_edges + 255) / 256, 256, 0, stream>>>(ei, et, cnt,
                                                           n_edges);
    rgcn_gemm<false><<<gemmBlocks, 256, 0, stream>>>(
        x, W, Wr, bias, nullptr, nullptr, out, n_nodes, tiles);
    edge_mm<<<edgeBlocks, 256, 0, stream>>>(x, W, ei, et, cnt, out, n_edges);
    relu_inplace<<<2048, 256, 0, stream>>>(out, (size_t)n_nodes * DOUT);
  }
}